// OGCA_MMDLoss_75462575391023
// MI455X (gfx1250) — compile-verified
//
#include <hip/hip_runtime.h>
#include <hip/hip_bf16.h>
#include <math.h>

typedef __attribute__((ext_vector_type(2))) float v2f;
typedef __attribute__((ext_vector_type(8))) float v8f;

#define Nn 8192
#define Mm 8192
#define Dd 128
#define EPSI 0.05f
#define SINK_ITERS 30
#define LOG_A (-9.010913347f)   /* -ln(8192) */
#define LOG_B (-9.010913347f)

#define TILE 128
#define LDSS 132   /* padded LDS row stride: 4-bank advance per row -> conflict-free frag loads */

__device__ __forceinline__ float ms_rbf_from_dot(float dot, float nx, float ny) {
    float d2 = nx + ny - 2.0f * dot;
    d2 = fmaxf(d2, 0.0f);
    float k = __expf(-0.5f        * d2);   // sigma = 1
    k += __expf(-0.125f           * d2);   // sigma = 2
    k += __expf(-0.03125f         * d2);   // sigma = 4
    k += __expf(-0.0078125f       * d2);   // sigma = 8
    k += __expf(-0.001953125f     * d2);   // sigma = 16
    return 0.2f * k;
}

// ---------------------------------------------------------------------------
// Fused GEMM + multiscale-RBF. mode 0: store Kxy + track global max.
//                              mode 1: accumulate double sum (for Kxx/Kyy means).
// Block = 256 threads = 8 waves; 128x128 output tile; WMMA f32 16x16x4, K=128.
// ---------------------------------------------------------------------------
__global__ __launch_bounds__(256)
void gemm_ms_kernel(const float* __restrict__ A, const float* __restrict__ B,
                    const float* __restrict__ anorm, const float* __restrict__ bnorm,
                    float* __restrict__ Kout, unsigned int* __restrict__ maxbits,
                    double* __restrict__ sumacc, int mode)
{
    __shared__ __align__(16) float Xs[TILE * LDSS];
    __shared__ __align__(16) float Ys[TILE * LDSS];

    const int t   = threadIdx.x;
    const int ib  = blockIdx.y, jb = blockIdx.x;
    const int row0 = ib * TILE, col0 = jb * TILE;

    // Cooperative load of X tile (128x128) and Y tile (128x128) into padded LDS.
    {
        const float4* Ag = (const float4*)(A + (size_t)row0 * Dd);
        const float4* Bg = (const float4*)(B + (size_t)col0 * Dd);
        #pragma unroll
        for (int it = 0; it < 16; ++it) {
            int fi = t + 256 * it;        // 0..4095 float4s
            int r  = fi >> 5;             // 32 float4 per row
            int c4 = fi & 31;
            float4 va = Ag[r * 32 + c4];
            float4 vb = Bg[r * 32 + c4];
            int base = r * LDSS + c4 * 4;
            Xs[base + 0] = va.x; Xs[base + 1] = va.y; Xs[base + 2] = va.z; Xs[base + 3] = va.w;
            Ys[base + 0] = vb.x; Ys[base + 1] = vb.y; Ys[base + 2] = vb.z; Ys[base + 3] = vb.w;
        }
    }
    __syncthreads();

    const int wave = t >> 5;      // 0..7 : 16-row strip of the 128-row tile
    const int lane = t & 31;
    const int lr   = lane & 15;   // M (A) / N (B) index within 16
    const int hi   = lane >> 4;   // selects K pair {0,1} vs {2,3}

    v8f acc[8];
    #pragma unroll
    for (int nt = 0; nt < 8; ++nt)
        #pragma unroll
        for (int r = 0; r < 8; ++r) acc[nt][r] = 0.0f;

    const float* arow = &Xs[(wave * 16 + lr) * LDSS];
    #pragma unroll 4
    for (int kb = 0; kb < Dd; kb += 4) {
        const int ko = kb + hi * 2;
        v2f a; a.x = arow[ko]; a.y = arow[ko + 1];
        #pragma unroll
        for (int nt = 0; nt < 8; ++nt) {
            const float* brow = &Ys[(nt * 16 + lr) * LDSS];
            v2f b; b.x = brow[ko]; b.y = brow[ko + 1];
            acc[nt] = __builtin_amdgcn_wmma_f32_16x16x4_f32(
                false, a, false, b, (short)0, acc[nt], false, false);
        }
    }

    // Epilogue: C/D layout -> VGPR r holds (M = r + 8*hi, N = lr) of each 16x16 subtile.
    float  tmax = 0.0f;
    double tsum = 0.0;
    #pragma unroll
    for (int nt = 0; nt < 8; ++nt) {
        #pragma unroll
        for (int r = 0; r < 8; ++r) {
            int gi = row0 + wave * 16 + r + 8 * hi;
            int gj = col0 + nt * 16 + lr;
            float kv = ms_rbf_from_dot(acc[nt][r], anorm[gi], bnorm[gj]);
            if (mode == 0) {
                Kout[(size_t)gi * Mm + gj] = kv;
                tmax = fmaxf(tmax, kv);
            } else {
                tsum += (double)kv;
            }
        }
    }

    __syncthreads();   // LDS tiles dead; reuse for block reduction
    if (mode == 0) {
        Xs[t] = tmax;
        __syncthreads();
        for (int s = 128; s > 0; s >>= 1) {
            if (t < s) Xs[t] = fmaxf(Xs[t], Xs[t + s]);
            __syncthreads();
        }
        if (t == 0) atomicMax(maxbits, __float_as_uint(Xs[0]));  // kv > 0 -> bit-monotonic
    } else {
        double* Sd = (double*)Ys;
        Sd[t] = tsum;
        __syncthreads();
        for (int s = 128; s > 0; s >>= 1) {
            if (t < s) Sd[t] += Sd[t + s];
            __syncthreads();
        }
        if (t == 0) atomicAdd(sumacc, Sd[0]);
    }
}

// ---------------------------------------------------------------------------
__global__ void rownorm_kernel(const float* __restrict__ X, float* __restrict__ nrm, int n)
{
    int i = blockIdx.x * blockDim.x + threadIdx.x;
    if (i >= n) return;
    const float4* row = (const float4*)(X + (size_t)i * Dd);
    float s = 0.0f;
    #pragma unroll
    for (int k = 0; k < Dd / 4; ++k) {
        float4 v = row[k];
        s += v.x * v.x + v.y * v.y + v.z * v.z + v.w * v.w;
    }
    nrm[i] = s;
}

__global__ void init_kernel(float* __restrict__ log_u, float* __restrict__ log_v,
                            unsigned int* __restrict__ maxbits, double* __restrict__ sums)
{
    int i = blockIdx.x * blockDim.x + threadIdx.x;
    if (i < Nn) log_u[i] = 0.0f;
    if (i < Mm) log_v[i] = 0.0f;
    if (i == 0) { *maxbits = 0u; sums[0] = 0.0; sums[1] = 0.0; sums[2] = 0.0; }
}

// log_u[i] = log_a - LSE_j( (K[i,j]-Kmax)/eps + log_v[j] ); one wave32 per row.
__global__ __launch_bounds__(256)
void row_lse_kernel(const float* __restrict__ K, const float* __restrict__ log_v,
                    float* __restrict__ log_u, const unsigned int* __restrict__ maxbits)
{
    const float kmax   = __uint_as_float(*maxbits);
    const float invEps = 1.0f / EPSI;
    const int wave = threadIdx.x >> 5, lane = threadIdx.x & 31;
    const int i = blockIdx.x * 8 + wave;
    const float* krow = K + (size_t)i * Mm;

    float m = -3.0e38f, s = 0.0f;
    for (int j = lane; j < Mm; j += 32) {
        float v = (krow[j] - kmax) * invEps + log_v[j];
        if (v > m) { s = s * __expf(m - v) + 1.0f; m = v; }
        else       { s += __expf(v - m); }
    }
    #pragma unroll
    for (int off = 16; off > 0; off >>= 1) {
        float mo = __shfl_xor(m, off, 32);
        float so = __shfl_xor(s, off, 32);
        float mn = fmaxf(m, mo);
        s = s * __expf(m - mn) + so * __expf(mo - mn);
        m = mn;
    }
    if (lane == 0) log_u[i] = LOG_A - (m + __logf(s));
}

// log_v[j] = log_b - LSE_i( (K[i,j]-Kmax)/eps + log_u[i] ); 64 cols/block, 4-way row split.
__global__ __launch_bounds__(256)
void col_lse_kernel(const float* __restrict__ K, const float* __restrict__ log_u,
                    float* __restrict__ log_v, const unsigned int* __restrict__ maxbits)
{
    const float kmax   = __uint_as_float(*maxbits);
    const float invEps = 1.0f / EPSI;
    const int jl  = threadIdx.x & 63;
    const int seg = threadIdx.x >> 6;          // 0..3
    const int j   = blockIdx.x * 64 + jl;

    float m = -3.0e38f, s = 0.0f;
    const int i0 = seg * (Nn / 4), i1 = i0 + (Nn / 4);
    for (int i = i0; i < i1; ++i) {
        float v = (K[(size_t)i * Mm + j] - kmax) * invEps + log_u[i];
        if (v > m) { s = s * __expf(m - v) + 1.0f; m = v; }
        else       { s += __expf(v - m); }
    }
    __shared__ float sm[4][64];
    __shared__ float ss[4][64];
    sm[seg][jl] = m; ss[seg][jl] = s;
    __syncthreads();
    if (seg == 0) {
        #pragma unroll
        for (int q = 1; q < 4; ++q) {
            float mo = sm[q][jl], so = ss[q][jl];
            float mn = fmaxf(m, mo);
            s = s * __expf(m - mn) + so * __expf(mo - mn);
            m = mn;
        }
        log_v[j] = LOG_B - (m + __logf(s));
    }
}

// sumGK += sum_ij exp(log_u[i] + (K-Kmax)/eps + log_v[j]) * K[i,j]
__global__ __launch_bounds__(256)
void final_reduce_kernel(const float* __restrict__ K, const float* __restrict__ log_u,
                         const float* __restrict__ log_v, const unsigned int* __restrict__ maxbits,
                         double* __restrict__ sumGK)
{
    const float kmax   = __uint_as_float(*maxbits);
    const float invEps = 1.0f / EPSI;
    const size_t total  = (size_t)Nn * Mm;
    const size_t stride = (size_t)gridDim.x * blockDim.x;
    double acc = 0.0;
    for (size_t p = (size_t)blockIdx.x * blockDim.x + threadIdx.x; p < total; p += stride) {
        int i = (int)(p >> 13);     // / 8192
        int j = (int)(p & 8191);
        float k = K[p];
        float g = __expf(log_u[i] + (k - kmax) * invEps + log_v[j]);
        acc += (double)(g * k);
    }
    __shared__ double sd[256];
    sd[threadIdx.x] = acc;
    __syncthreads();
    for (int s = 128; s > 0; s >>= 1) {
        if (threadIdx.x < s) sd[threadIdx.x] += sd[threadIdx.x + s];
        __syncthreads();
    }
    if (threadIdx.x == 0) atomicAdd(sumGK, sd[0]);
}

__global__ void combine_kernel(const double* __restrict__ sums, float* __restrict__ out)
{
    double kxx = sums[0] / ((double)Nn * (double)Nn);
    double kyy = sums[1] / ((double)Mm * (double)Mm);
    out[0] = (float)(kxx + kyy - 2.0 * sums[2]);
}

// ---------------------------------------------------------------------------
extern "C" void kernel_launch(void* const* d_in, const int* in_sizes, int n_in,
                              void* d_out, int out_size, void* d_ws, size_t ws_size,
                              hipStream_t stream)
{
    (void)in_sizes; (void)n_in; (void)out_size; (void)ws_size;
    const float* X = (const float*)d_in[0];
    const float* Y = (const float*)d_in[1];
    float* out = (float*)d_out;

    char* ws = (char*)d_ws;
    size_t o = 0;
    float* Kxy   = (float*)(ws + o); o += (size_t)Nn * Mm * sizeof(float);
    float* xnorm = (float*)(ws + o); o += (size_t)Nn * sizeof(float);
    float* ynorm = (float*)(ws + o); o += (size_t)Mm * sizeof(float);
    float* log_u = (float*)(ws + o); o += (size_t)Nn * sizeof(float);
    float* log_v = (float*)(ws + o); o += (size_t)Mm * sizeof(float);
    o = (o + 255) & ~(size_t)255;
    double*       sums    = (double*)(ws + o); o += 3 * sizeof(double);
    unsigned int* maxbits = (unsigned int*)(ws + o);

    init_kernel<<<(Nn + 255) / 256, 256, 0, stream>>>(log_u, log_v, maxbits, sums);
    rownorm_kernel<<<(Nn + 255) / 256, 256, 0, stream>>>(X, xnorm, Nn);
    rownorm_kernel<<<(Mm + 255) / 256, 256, 0, stream>>>(Y, ynorm, Mm);

    dim3 grid(Mm / TILE, Nn / TILE);
    gemm_ms_kernel<<<grid, 256, 0, stream>>>(X, Y, xnorm, ynorm, Kxy, maxbits, nullptr, 0);
    gemm_ms_kernel<<<grid, 256, 0, stream>>>(X, X, xnorm, xnorm, nullptr, nullptr, &sums[0], 1);
    gemm_ms_kernel<<<grid, 256, 0, stream>>>(Y, Y, ynorm, ynorm, nullptr, nullptr, &sums[1], 1);

    for (int it = 0; it < SINK_ITERS; ++it) {
        row_lse_kernel<<<Nn / 8, 256, 0, stream>>>(Kxy, log_v, log_u, maxbits);
        col_lse_kernel<<<Mm / 64, 256, 0, stream>>>(Kxy, log_u, log_v, maxbits);
    }

    final_reduce_kernel<<<4096, 256, 0, stream>>>(Kxy, log_u, log_v, maxbits, &sums[2]);
    combine_kernel<<<1, 1, 0, stream>>>(sums, out);
}